// QuinticHermiteSpline_87540023427507
// MI455X (gfx1250) — compile-verified
//
#include <hip/hip_runtime.h>
#include <stdint.h>

#define NK   1024            // number of knots
#define NTBL 1024            // interval table entries (1023 intervals, padded)
#define BLOCK 256
#define STAGE_ITERS ((2 * NTBL) / BLOCK)   // 8 x b128 per thread = 32 KB staged

typedef float v4f __attribute__((ext_vector_type(4)));

// ---------------------------------------------------------------------------
// Pre-pass: per-interval quintic Hermite coefficients, packed as 2 float4s.
//   A[k] = { x_l, 1/h, yl, h*dyl }
//   B[k] = { 0.5*h^2*ddyl, c3, c4, c5 }
// Hot kernel then needs only 2 x ds_load_b128 + 5 fma per query.
// ---------------------------------------------------------------------------
__global__ __launch_bounds__(BLOCK)
void spline_coeff_kernel(const float* __restrict__ knots,
                         const float* __restrict__ fv,     // [3][NK]
                         v4f* __restrict__ tbl)            // [2*NTBL]
{
    int k = blockIdx.x * BLOCK + threadIdx.x;
    if (k >= NTBL) return;
    int kk = (k < NK - 1) ? k : (NK - 2);                  // pad last entry

    const float* y   = fv;
    const float* dy  = fv + NK;
    const float* ddy = fv + 2 * NK;

    float x_l  = knots[kk];
    float h    = knots[kk + 1] - x_l;
    float yl   = y[kk],   yr   = y[kk + 1];
    float dyl  = dy[kk],  dyr  = dy[kk + 1];
    float ddyl = ddy[kk], ddyr = ddy[kk + 1];

    float dY = yr - yl;
    float h2 = 0.5f * h * h;
    float c5 =   6.0f * dY - 3.0f * h * (dyl + dyr)           + h2 * (ddyr - ddyl);
    float c4 = -15.0f * dY + h * (8.0f * dyl + 7.0f * dyr)    - h2 * (2.0f * ddyr - 3.0f * ddyl);
    float c3 =  10.0f * dY - 2.0f * h * (3.0f * dyl + 2.0f * dyr) + h2 * (ddyr - 3.0f * ddyl);

    v4f A = { x_l, 1.0f / h, yl, h * dyl };
    v4f B = { h2 * ddyl, c3, c4, c5 };
    tbl[k]        = A;
    tbl[NTBL + k] = B;
}

// ---------------------------------------------------------------------------
// Helpers
// ---------------------------------------------------------------------------
__device__ __forceinline__ int interval_index(float xq) {
    int k = (int)(xq * (float)(NK - 1));   // uniform knots: direct index
    k = (k < 0) ? 0 : k;
    k = (k > NK - 2) ? (NK - 2) : k;
    return k;
}

// Array-reference parameter keeps addrspace(3) so gathers lower to ds_load_b128.
__device__ __forceinline__ float eval_poly(float xq, const v4f (&tbl)[2 * NTBL]) {
    int k = interval_index(xq);
    v4f A = tbl[k];
    v4f B = tbl[NTBL + k];
    float t = (xq - A.x) * A.y;
    float v = fmaf(B.w, t, B.z);   // c5*t + c4
    v = fmaf(v, t, B.y);           // + c3
    v = fmaf(v, t, B.x);           // + h2*ddyl
    v = fmaf(v, t, A.w);           // + h*dyl
    return fmaf(v, t, A.z);        // + yl
}

__device__ __forceinline__ v4f eval_poly4(v4f xv, const v4f (&tbl)[2 * NTBL]) {
    v4f r;
    r.x = eval_poly(xv.x, tbl);
    r.y = eval_poly(xv.y, tbl);
    r.z = eval_poly(xv.z, tbl);
    r.w = eval_poly(xv.w, tbl);
    return r;
}

// ---------------------------------------------------------------------------
// Hot kernel: async-stage 32 KB coefficient table into LDS (gfx1250
// global_load_async_to_lds_b128 / ASYNCcnt path), then stream queries as
// non-temporal b128 and gather coefficients from LDS.  NT hints keep the
// 128 MB one-touch stream from churning L2; the table stays L2-resident.
// ---------------------------------------------------------------------------
__global__ __launch_bounds__(BLOCK)
void spline_eval_kernel(const float* __restrict__ xin,
                        const v4f* __restrict__ tbl_g,
                        float* __restrict__ out, int n)
{
    __shared__ v4f tbl[2 * NTBL];   // 32 KB

    // Low 32 bits of the flat shared address = wave-relative LDS byte offset.
    uint32_t lds_base = (uint32_t)(uintptr_t)(&tbl[0]);
#pragma unroll
    for (int j = 0; j < STAGE_ITERS; ++j) {
        uint32_t e     = (uint32_t)threadIdx.x + (uint32_t)(BLOCK * j);
        uint32_t laddr = lds_base + e * 16u;
        uint64_t gaddr = (uint64_t)(uintptr_t)(tbl_g + e);
        asm volatile("global_load_async_to_lds_b128 %0, %1, off"
                     :: "v"(laddr), "v"(gaddr) : "memory");
    }
    asm volatile("s_wait_asynccnt 0x0" ::: "memory");  // this wave's asyncs done
    __syncthreads();                                   // all waves' data visible

    const v4f* __restrict__ x4 = (const v4f*)xin;
    v4f* __restrict__ o4 = (v4f*)out;
    int tid      = blockIdx.x * BLOCK + threadIdx.x;
    int nthreads = (int)(gridDim.x * blockDim.x);

    int n4 = n >> 2;   // float4 count
    int n8 = n >> 3;   // pairs of float4

    // Main loop: 8 queries / iter -> 2 independent b128 loads + 16 ds_load_b128
    // in flight ahead of the dependent FMA chains.
    for (int i = tid; i < n8; i += nthreads) {
        v4f xa = __builtin_nontemporal_load(x4 + 2 * i);
        v4f xb = __builtin_nontemporal_load(x4 + 2 * i + 1);
        v4f ra = eval_poly4(xa, tbl);
        v4f rb = eval_poly4(xb, tbl);
        __builtin_nontemporal_store(ra, o4 + 2 * i);
        __builtin_nontemporal_store(rb, o4 + 2 * i + 1);
    }
    // Remaining single float4s (empty when n % 8 == 0).
    for (int i = 2 * n8 + tid; i < n4; i += nthreads) {
        v4f xv = __builtin_nontemporal_load(x4 + i);
        v4f r  = eval_poly4(xv, tbl);
        __builtin_nontemporal_store(r, o4 + i);
    }
    // Scalar tail (empty when n % 4 == 0; n is 2^24).
    for (int i = (n4 << 2) + tid; i < n; i += nthreads) {
        out[i] = eval_poly(xin[i], tbl);
    }
}

// ---------------------------------------------------------------------------
// Fallback (no workspace): compute everything from global memory / L2.
// ---------------------------------------------------------------------------
__global__ __launch_bounds__(BLOCK)
void spline_eval_global_kernel(const float* __restrict__ xin,
                               const float* __restrict__ knots,
                               const float* __restrict__ fv,
                               float* __restrict__ out, int n)
{
    const float* y   = fv;
    const float* dy  = fv + NK;
    const float* ddy = fv + 2 * NK;
    int stride = (int)(gridDim.x * blockDim.x);
    for (int i = blockIdx.x * BLOCK + threadIdx.x; i < n; i += stride) {
        float xq = xin[i];
        int k = interval_index(xq);
        float x_l  = knots[k];
        float h    = knots[k + 1] - x_l;
        float yl   = y[k],   yr   = y[k + 1];
        float dyl  = dy[k],  dyr  = dy[k + 1];
        float ddyl = ddy[k], ddyr = ddy[k + 1];
        float dY = yr - yl;
        float h2 = 0.5f * h * h;
        float c5 =   6.0f * dY - 3.0f * h * (dyl + dyr)           + h2 * (ddyr - ddyl);
        float c4 = -15.0f * dY + h * (8.0f * dyl + 7.0f * dyr)    - h2 * (2.0f * ddyr - 3.0f * ddyl);
        float c3 =  10.0f * dY - 2.0f * h * (3.0f * dyl + 2.0f * dyr) + h2 * (ddyr - 3.0f * ddyl);
        float t = (xq - x_l) / h;
        float v = fmaf(c5, t, c4);
        v = fmaf(v, t, c3);
        v = fmaf(v, t, h2 * ddyl);
        v = fmaf(v, t, h * dyl);
        out[i] = fmaf(v, t, yl);
    }
}

// ---------------------------------------------------------------------------
extern "C" void kernel_launch(void* const* d_in, const int* in_sizes, int n_in,
                              void* d_out, int out_size, void* d_ws, size_t ws_size,
                              hipStream_t stream)
{
    const float* x     = (const float*)d_in[0];   // [N] queries
    const float* knots = (const float*)d_in[1];   // [1024]
    const float* fv    = (const float*)d_in[2];   // [3][1024]
    float* out = (float*)d_out;
    int n = in_sizes[0];

    if (d_ws != nullptr && ws_size >= (size_t)(2 * NTBL) * sizeof(v4f)) {
        v4f* tbl = (v4f*)d_ws;
        spline_coeff_kernel<<<(NTBL + BLOCK - 1) / BLOCK, BLOCK, 0, stream>>>(knots, fv, tbl);
        spline_eval_kernel<<<2048, BLOCK, 0, stream>>>(x, tbl, out, n);
    } else {
        spline_eval_global_kernel<<<2048, BLOCK, 0, stream>>>(x, knots, fv, out, n);
    }
}